// RWKV_6_9500467659408
// MI455X (gfx1250) — compile-verified
//
#include <hip/hip_runtime.h>

typedef __attribute__((ext_vector_type(2))) float v2f;
typedef __attribute__((ext_vector_type(8))) float v8f;
typedef __attribute__((ext_vector_type(4))) int   v4i;

#define Bsz 8
#define Hn  32
#define Tn  2048
#define Kd  64
#define Vd  64
#define CL  16   // chunk length

// ---- CDNA5 async global->LDS copy (ASYNCcnt path), with portable fallback ----
#if defined(__has_builtin)
#  if __has_builtin(__builtin_amdgcn_global_load_async_to_lds_b128)
#    define USE_ASYNC_LDS 1
#  endif
#endif

__device__ __forceinline__ void copy16_to_lds(const float4* g, float4* l) {
#if defined(USE_ASYNC_LDS)
    // signature (from hipcc diagnostic): (v4i AS1*, v4i AS3*, imm offset, imm cpol)
    __builtin_amdgcn_global_load_async_to_lds_b128(
        (v4i __attribute__((address_space(1)))*)g,
        (v4i __attribute__((address_space(3)))*)l,
        /*offset=*/0, /*cpol=*/0);
#else
    *l = *g;
#endif
}

__device__ __forceinline__ void async_copies_wait() {
#if defined(USE_ASYNC_LDS)
#  if __has_builtin(__builtin_amdgcn_s_wait_asynccnt)
    __builtin_amdgcn_s_wait_asynccnt(0);
#  else
    asm volatile("s_wait_asynccnt 0" ::: "memory");
#  endif
#endif
}

__device__ __forceinline__ v8f wmma_f32(v2f a, v2f b, v8f c) {
    // V_WMMA_F32_16X16X4_F32 : D(16x16,f32) = A(16x4,f32) x B(4x16,f32) + C
    return __builtin_amdgcn_wmma_f32_16x16x4_f32(
        /*neg_a=*/false, a, /*neg_b=*/false, b,
        /*c_mod=*/(short)0, c, /*reuse_a=*/false, /*reuse_b=*/false);
}

__global__ __launch_bounds__(128)
void rwkv6_chunk_kernel(const float* __restrict__ r,
                        const float* __restrict__ k,
                        const float* __restrict__ v,
                        const float* __restrict__ w,
                        const float* __restrict__ u,
                        const float* __restrict__ h0,
                        float* __restrict__ o,
                        float* __restrict__ hout)
{
    // double-buffered raw chunk data (q~,k~ transformed in place)
    __shared__ float sQ [2][CL*Kd];
    __shared__ float sK [2][CL*Kd];
    __shared__ float sV [2][CL*Vd];
    __shared__ float sWr[2][CL*Kd];
    __shared__ float sKh[CL*Kd];        // k^ (decay-to-chunk-end)
    __shared__ float sP [(CL+1)*Kd];    // cumulative log-decay, P[0]=0..P[16]
    __shared__ float sH [Kd*Vd];        // running state (16 KB)
    __shared__ float sS [4*CL*CL];      // per-wave masked S
    __shared__ float sDiag[CL];
    __shared__ float sU [Kd];
    __shared__ float saL[Kd];           // e^{P_L}

    const int tid  = threadIdx.x;
    const int bh   = blockIdx.x;
    const int h    = bh % Hn;
    const int lane = tid & 31;
    const int wv   = tid >> 5;          // wave id: owns V-cols [16*wv, 16*wv+16)
    const int m    = lane & 15;         // A row / B,C,D column within tile
    const int hi   = lane >> 4;
    const int koff = hi << 1;           // inner-K sub-offset per ISA layout

    const size_t base = (size_t)bh * Tn * Kd;

    // issue chunk copy (async on CDNA5; each 16x64 chunk is contiguous 4 KB)
    auto load_chunk = [&](int c0, int p) {
        const size_t cb = base + (size_t)c0 * Kd;
        const float4* rv4 = (const float4*)(r + cb);
        const float4* kv4 = (const float4*)(k + cb);
        const float4* vv4 = (const float4*)(v + cb);
        const float4* wv4 = (const float4*)(w + cb);
        float4* q4 = (float4*)sQ[p];  float4* k4 = (float4*)sK[p];
        float4* v4 = (float4*)sV[p];  float4* w4 = (float4*)sWr[p];
        #pragma unroll
        for (int j = 0; j < 2; ++j) {
            int idx = tid + j*128;
            copy16_to_lds(rv4 + idx, q4 + idx);
            copy16_to_lds(kv4 + idx, k4 + idx);
            copy16_to_lds(vv4 + idx, v4 + idx);
            copy16_to_lds(wv4 + idx, w4 + idx);
        }
    };

    // ---- init state + u, and prefetch first chunk ----
    {
        const float4* h0v = (const float4*)(h0 + (size_t)bh * Kd * Vd);
        float4* sHv = (float4*)sH;
        #pragma unroll
        for (int j = 0; j < 8; ++j) sHv[tid*8 + j] = h0v[tid*8 + j];
        if (tid < Kd) sU[tid] = u[h*Kd + tid];
    }
    load_chunk(0, 0);

    int p = 0;
    for (int c0 = 0; c0 < Tn; c0 += CL, p ^= 1) {
        async_copies_wait();        // this wave's copies into buf[p] complete
        __syncthreads();            // all waves' copies complete; prev compute done

        // prefetch next chunk into the other buffer while we compute on buf[p]
        if (c0 + CL < Tn) load_chunk(c0 + CL, p ^ 1);

        float* Q  = sQ[p];
        float* Kt = sK[p];
        float* Vc = sV[p];
        float* Wr = sWr[p];

        // ---- cumulative log-decay (1 thread/channel) + diag term (1 thread/row) ----
        if (tid < Kd) {
            float run = 0.f;
            sP[tid] = 0.f;
            #pragma unroll
            for (int i = 0; i < CL; ++i) {
                run += Wr[i*Kd + tid];
                sP[(i+1)*Kd + tid] = run;
            }
            saL[tid] = __expf(run);
        } else if (tid < Kd + CL) {
            int i = tid - Kd;
            float s = 0.f;
            #pragma unroll
            for (int c = 0; c < Kd; ++c)
                s += Q[i*Kd + c] * sU[c] * Kt[i*Kd + c];   // raw q,k
            sDiag[i] = s;
        }
        __syncthreads();

        // ---- transforms: q~, k~, k^ (8 elements/thread, same time-row) ----
        {
            const int e0 = tid * 8;
            const int i  = e0 >> 6;
            #pragma unroll
            for (int j = 0; j < 8; ++j) {
                int e = e0 + j, c = e & 63;
                float pi  = sP[i*Kd + c];
                float pi1 = sP[(i+1)*Kd + c];
                float pl  = sP[CL*Kd + c];
                Q[e] *= __expf(pi);
                float kr = Kt[e];
                Kt[e]  = kr * __expf(-pi1);
                sKh[e] = kr * __expf(pl - pi1);
            }
        }
        __syncthreads();

        // ---- Phase B: S = q~ @ k~^T (16x16, inner K=64), per-wave redundant ----
        v8f accS = {0,0,0,0,0,0,0,0};
        #pragma unroll
        for (int k0 = 0; k0 < Kd; k0 += 4) {
            v2f a = { Q [m*Kd + k0 + koff], Q [m*Kd + k0 + koff + 1] };
            v2f b = { Kt[m*Kd + k0 + koff], Kt[m*Kd + k0 + koff + 1] };
            accS = wmma_f32(a, b, accS);
        }
        {   // causal mask + diagonal, store to this wave's S region
            float* Sp = sS + wv*(CL*CL);
            #pragma unroll
            for (int i = 0; i < 8; ++i) {
                int row = i + hi*8;
                float sv = accS[i];
                sv = (m > row) ? 0.f : ((m == row) ? sDiag[row] : sv);
                Sp[row*CL + m] = sv;
            }
        }

        // ---- Phase C: O[:,slice] = S @ Vc[:,slice] + q~ @ H[:,slice] ----
        v8f accO = {0,0,0,0,0,0,0,0};
        {
            const float* Sp = sS + wv*(CL*CL);
            #pragma unroll
            for (int k0 = 0; k0 < CL; k0 += 4) {
                v2f a = { Sp[m*CL + k0 + koff], Sp[m*CL + k0 + koff + 1] };
                v2f b = { Vc[(k0+koff  )*Vd + wv*16 + m],
                          Vc[(k0+koff+1)*Vd + wv*16 + m] };
                accO = wmma_f32(a, b, accO);
            }
            #pragma unroll
            for (int k0 = 0; k0 < Kd; k0 += 4) {
                v2f a = { Q [m*Kd + k0 + koff], Q [m*Kd + k0 + koff + 1] };
                v2f b = { sH[(k0+koff  )*Vd + wv*16 + m],
                          sH[(k0+koff+1)*Vd + wv*16 + m] };
                accO = wmma_f32(a, b, accO);
            }
        }
        {
            float* ob = o + base + (size_t)c0 * Vd;
            #pragma unroll
            for (int i = 0; i < 8; ++i) {
                int row = i + hi*8;
                ob[row*Vd + wv*16 + m] = accO[i];
            }
        }

        // ---- Phase D: H[:,slice] = diag(e^{P_L})*H[:,slice] + k^T @ Vc[:,slice] ----
        #pragma unroll
        for (int mt = 0; mt < 4; ++mt) {
            v8f accH;
            #pragma unroll
            for (int i = 0; i < 8; ++i) {
                int rr = mt*16 + i + hi*8;
                accH[i] = saL[rr] * sH[rr*Vd + wv*16 + m];
            }
            #pragma unroll
            for (int k0 = 0; k0 < CL; k0 += 4) {
                v2f a = { sKh[(k0+koff  )*Kd + mt*16 + m],
                          sKh[(k0+koff+1)*Kd + mt*16 + m] };
                v2f b = { Vc[(k0+koff  )*Vd + wv*16 + m],
                          Vc[(k0+koff+1)*Vd + wv*16 + m] };
                accH = wmma_f32(a, b, accH);
            }
            #pragma unroll
            for (int i = 0; i < 8; ++i) {
                int rr = mt*16 + i + hi*8;
                sH[rr*Vd + wv*16 + m] = accH[i];
            }
        }
        // no trailing barrier: next iteration's data-ready barrier orders
        // buffer reuse, and each wave exclusively owns its H column slice.
    }

    __syncthreads();
    // ---- final state out ----
    {
        float4* hov = (float4*)(hout + (size_t)bh * Kd * Vd);
        float4* sHv = (float4*)sH;
        #pragma unroll
        for (int j = 0; j < 8; ++j) hov[tid*8 + j] = sHv[tid*8 + j];
    }
}

extern "C" void kernel_launch(void* const* d_in, const int* in_sizes, int n_in,
                              void* d_out, int out_size, void* d_ws, size_t ws_size,
                              hipStream_t stream) {
    const float* r  = (const float*)d_in[0];
    const float* k  = (const float*)d_in[1];
    const float* v  = (const float*)d_in[2];
    const float* w  = (const float*)d_in[3];
    const float* u  = (const float*)d_in[4];
    const float* h0 = (const float*)d_in[5];
    float* o    = (float*)d_out;
    float* hout = o + (size_t)Bsz * Hn * Tn * Vd;
    rwkv6_chunk_kernel<<<Bsz * Hn, 128, 0, stream>>>(r, k, v, w, u, h0, o, hout);
}